// MuonMamba_57964878627188
// MI455X (gfx1250) — compile-verified
//
#include <hip/hip_runtime.h>
#include <cstdint>
#include <cstddef>

#define D_MODEL   768
#define N_LAYERS  2
#define D_STATE   16
#define D_CONV    4
#define D_INNER   1536
#define DT_RANK   48
#define SEQ_LEN   512
#define BATCH_N   2
#define NTOK      (BATCH_N * SEQ_LEN)      /* 1024 */
#define XDBL_C    (DT_RANK + 2 * D_STATE)  /* 80 */

#define BETA_C    0.9f
#define NS_A      3.4445f
#define NS_B      (-4.775f)
#define NS_C      2.0315f
#define EPS_C     1e-5f

typedef float v2f __attribute__((ext_vector_type(2)));
typedef float v8f __attribute__((ext_vector_type(8)));

static __device__ __forceinline__ v8f wmma_f32_k4(v2f a, v2f b, v8f c) {
  // D = A(16x4,f32) * B(4x16,f32) + C(16x16,f32)
  return __builtin_amdgcn_wmma_f32_16x16x4_f32(false, a, false, b, (short)0, c,
                                               false, false);
}

// ---------------------------------------------------------------------------
// RMSNorm: one block per row of D floats.
// ---------------------------------------------------------------------------
__global__ void rmsnorm_kernel(const float* __restrict__ x,
                               const float* __restrict__ w,
                               float* __restrict__ out, int D) {
  __shared__ float red[8];
  const int row = blockIdx.x;
  const float* xr = x + (size_t)row * D;
  float ss = 0.f;
  for (int i = threadIdx.x; i < D; i += blockDim.x) {
    float v = xr[i];
    ss += v * v;
  }
  #pragma unroll
  for (int off = 16; off > 0; off >>= 1) ss += __shfl_down(ss, off);
  const int lane = threadIdx.x & 31, wid = threadIdx.x >> 5;
  if (lane == 0) red[wid] = ss;
  __syncthreads();
  if (threadIdx.x == 0) {
    float t = 0.f;
    for (int i = 0; i < (int)(blockDim.x >> 5); ++i) t += red[i];
    red[0] = rsqrtf(t / (float)D + EPS_C);
  }
  __syncthreads();
  const float scale = red[0];
  float* orow = out + (size_t)row * D;
  for (int i = threadIdx.x; i < D; i += blockDim.x)
    orow[i] = xr[i] * scale * w[i];
}

// ---------------------------------------------------------------------------
// Generic f32 WMMA GEMM: C[M,N] = A[M,K] * W[N,K]^T
// grid = (ceil(N/64), M/16), block = 128 (4 waves, one 16x16 N-tile each).
// ---------------------------------------------------------------------------
__global__ void gemm_wmma_f32(const float* __restrict__ A, int lda,
                              const float* __restrict__ W, int ldw,
                              float* __restrict__ C, int ldc, int N, int K) {
  const int lane = threadIdx.x & 31;
  const int wid  = threadIdx.x >> 5;
  const int hf   = lane >> 4;   // lane half: selects K pair
  const int r    = lane & 15;   // A row within tile / B column within tile
  const int ntile = blockIdx.x * 4 + wid;
  if (ntile * 16 >= N) return;
  const int row0 = blockIdx.y * 16;
  const int col0 = ntile * 16;

  const float* __restrict__ arow = A + (size_t)(row0 + r) * lda + 2 * hf;
  const float* __restrict__ wrow = W + (size_t)(col0 + r) * ldw + 2 * hf;

  v8f c = {};
  for (int k = 0; k < K; k += 4) {
    v2f a, b;
    a.x = arow[k]; a.y = arow[k + 1];
    b.x = wrow[k]; b.y = wrow[k + 1];
    c = wmma_f32_k4(a, b, c);
  }
  float* crow = C + (size_t)(row0 + hf * 8) * ldc + (col0 + r);
  #pragma unroll
  for (int i = 0; i < 8; ++i) crow[(size_t)i * ldc] = c[i];
}

// ---------------------------------------------------------------------------
// Depthwise causal conv (K=4) + bias + SiLU.  xi = xz[:, :D_INNER].
// ---------------------------------------------------------------------------
__global__ void conv_silu_kernel(const float* __restrict__ xz,
                                 const float* __restrict__ cw,
                                 const float* __restrict__ cb,
                                 float* __restrict__ xc) {
  const int idx = blockIdx.x * blockDim.x + threadIdx.x;
  if (idx >= NTOK * D_INNER) return;
  const int d  = idx % D_INNER;
  const int bl = idx / D_INNER;
  const int l  = bl % SEQ_LEN;
  const int b  = bl / SEQ_LEN;
  float acc = cb[d];
  #pragma unroll
  for (int j = 0; j < D_CONV; ++j) {
    const int ll = l - (D_CONV - 1) + j;
    if (ll >= 0)
      acc += xz[((size_t)(b * SEQ_LEN + ll)) * (2 * D_INNER) + d] * cw[d * D_CONV + j];
  }
  xc[idx] = acc / (1.f + __expf(-acc));  // silu
}

// ---------------------------------------------------------------------------
// In-place softplus(dt_raw + bias)
// ---------------------------------------------------------------------------
__global__ void softplus_kernel(float* __restrict__ dt,
                                const float* __restrict__ bias) {
  const int idx = blockIdx.x * blockDim.x + threadIdx.x;
  if (idx >= NTOK * D_INNER) return;
  const float v = dt[idx] + bias[idx % D_INNER];
  dt[idx] = (v > 20.f) ? v : log1pf(__expf(v));
}

// negA[d][n] = -exp(A_log[d][n])
__global__ void nega_kernel(const float* __restrict__ alog,
                            float* __restrict__ negA) {
  const int idx = blockIdx.x * blockDim.x + threadIdx.x;
  if (idx >= D_INNER * D_STATE) return;
  negA[idx] = -__expf(alog[idx]);
}

// ---------------------------------------------------------------------------
// Sequential scan with Newton-Schulz (1 step), one workgroup per batch.
// 512 threads = 16 waves; each thread owns 3 rows of h/v/negA in registers.
// v (1536x16) staged in LDS; Gram and NS transform via WMMA f32 16x16x4.
// ---------------------------------------------------------------------------
#define SCAN_THREADS 512
#define RPT 3                     /* D_INNER / SCAN_THREADS */
#define GRAM_KCH (D_INNER / 4 / 16) /* 24 k-chunks per wave */

__global__ void scan_kernel(const float* __restrict__ dtb,   // (NTOK, D_INNER)
                            const float* __restrict__ xdbl,  // (NTOK, 80)
                            const float* __restrict__ xc,    // (NTOK, D_INNER)
                            const float* __restrict__ negA,  // (D_INNER, 16)
                            float* __restrict__ ycore) {     // (NTOK, D_INNER)
  const int b    = blockIdx.x;
  const int tid  = threadIdx.x;
  const int lane = tid & 31;
  const int wid  = tid >> 5;
  const int hf   = lane >> 4;
  const int r    = lane & 15;

  extern __shared__ float smem[];
  float* vbuf  = smem;                    // D_INNER*16
  float* vub   = vbuf + D_INNER * D_STATE; // D_INNER*16
  float* gpart = vub + D_INNER * D_STATE;  // 16 waves * 256
  float* Pm    = gpart + 16 * 256;         // 16x16  (P matrix)
  float* Rm    = Pm + 256;                 // 16x16  (Gram, then reused)
  float* Am    = gpart;                    // reuse gpart[0..255] for A matrix
  float* bc    = Rm + 256;                 // 32: Bm[0..15], Cm[16..31]
  float* sc    = bc + 32;                  // scalars: inv_s, inv_s2

  const int d0 = tid * RPT;
  float h[RPT][D_STATE], v[RPT][D_STATE], nA[RPT][D_STATE], dts[RPT];
  #pragma unroll
  for (int rr = 0; rr < RPT; ++rr)
    #pragma unroll
    for (int n = 0; n < D_STATE; ++n) {
      h[rr][n] = 0.f;
      v[rr][n] = 0.f;
      nA[rr][n] = negA[(size_t)(d0 + rr) * D_STATE + n];
    }

  for (int t = 0; t < SEQ_LEN; ++t) {
    const size_t bl = (size_t)b * SEQ_LEN + t;

    // ---- stage 0: broadcast Bm / Cm for this step -------------------------
    if (tid < 32) bc[tid] = xdbl[bl * XDBL_C + DT_RANK + tid];
    __syncthreads();

    // ---- stage 1: v = beta*v + dt*xc*B ; write v to LDS -------------------
    #pragma unroll
    for (int rr = 0; rr < RPT; ++rr) {
      const int d = d0 + rr;
      dts[rr] = dtb[bl * D_INNER + d];
      const float dx = dts[rr] * xc[bl * D_INNER + d];
      #pragma unroll
      for (int n = 0; n < D_STATE; ++n) {
        const float vv = BETA_C * v[rr][n] + dx * bc[n];
        v[rr][n] = vv;
        vbuf[(size_t)d * D_STATE + n] = vv;
      }
    }
    __syncthreads();

    // ---- stage 2: Gram R = v^T v via WMMA (A == B, symmetric) -------------
    {
      v8f cg = {};
      const int kbase = wid * GRAM_KCH * 4 + 2 * hf;
      for (int i = 0; i < GRAM_KCH; ++i) {
        const int k = kbase + i * 4;
        v2f a;
        a.x = vbuf[(size_t)k * D_STATE + r];
        a.y = vbuf[(size_t)(k + 1) * D_STATE + r];
        cg = wmma_f32_k4(a, a, cg);
      }
      #pragma unroll
      for (int i = 0; i < 8; ++i)
        gpart[wid * 256 + (hf * 8 + i) * 16 + r] = cg[i];
    }
    __syncthreads();

    // reduce partial Grams -> R
    if (tid < 256) {
      float s = 0.f;
      #pragma unroll
      for (int w = 0; w < 16; ++w) s += gpart[w * 256 + tid];
      Rm[tid] = s;
    }
    __syncthreads();

    // fro = sqrt(trace(R));  s = fro + 1e-7
    if (tid == 0) {
      float tr = 0.f;
      #pragma unroll
      for (int i = 0; i < 16; ++i) tr += Rm[i * 17];
      const float s = sqrtf(tr) + 1e-7f;
      sc[0] = 1.f / s;
      sc[1] = 1.f / (s * s);
    }
    __syncthreads();

    // A = R / s^2  (into reused gpart[0..255])
    if (tid < 256) Am[tid] = Rm[tid] * sc[1];
    __syncthreads();

    // P = b*A + c*A@A
    if (tid < 256) {
      const int row = tid >> 4, col = tid & 15;
      float a2 = 0.f;
      #pragma unroll
      for (int k = 0; k < 16; ++k) a2 += Am[row * 16 + k] * Am[k * 16 + col];
      Pm[tid] = NS_B * Am[tid] + NS_C * a2;
    }
    __syncthreads();

    // ---- stage 3: vu = (a*v + v @ P^T) / s  via WMMA ----------------------
    {
      const float inv_s = sc[0];
      for (int mt = wid; mt < D_INNER / 16; mt += 16) {
        const int m0 = mt * 16;
        v8f cv;
        #pragma unroll
        for (int i = 0; i < 8; ++i)
          cv[i] = NS_A * vbuf[(size_t)(m0 + hf * 8 + i) * D_STATE + r];
        #pragma unroll
        for (int k = 0; k < 16; k += 4) {
          v2f a, bb;
          a.x  = vbuf[(size_t)(m0 + r) * D_STATE + k + 2 * hf];
          a.y  = vbuf[(size_t)(m0 + r) * D_STATE + k + 2 * hf + 1];
          bb.x = Pm[r * 16 + k + 2 * hf];       // B[k][n] = P^T[k][n] = P[n][k]
          bb.y = Pm[r * 16 + k + 2 * hf + 1];
          cv = wmma_f32_k4(a, bb, cv);
        }
        #pragma unroll
        for (int i = 0; i < 8; ++i)
          vub[(size_t)(m0 + hf * 8 + i) * D_STATE + r] = cv[i] * inv_s;
      }
    }
    __syncthreads();

    // ---- stage 4: h = exp(dt*(-expA)) * h + vu ; y = h . C ----------------
    #pragma unroll
    for (int rr = 0; rr < RPT; ++rr) {
      const int d = d0 + rr;
      float yacc = 0.f;
      #pragma unroll
      for (int n = 0; n < D_STATE; ++n) {
        const float dA = __expf(dts[rr] * nA[rr][n]);
        const float hh = dA * h[rr][n] + vub[(size_t)d * D_STATE + n];
        h[rr][n] = hh;
        yacc += hh * bc[16 + n];
      }
      ycore[bl * D_INNER + d] = yacc;
    }
    __syncthreads();
  }
}

// ---------------------------------------------------------------------------
// y = (ycore + D*xc) * silu(z)      z = xz[:, D_INNER:]
// ---------------------------------------------------------------------------
__global__ void combine_kernel(const float* __restrict__ ycore,
                               const float* __restrict__ xc,
                               const float* __restrict__ xz,
                               const float* __restrict__ Dp,
                               float* __restrict__ yact) {
  const int idx = blockIdx.x * blockDim.x + threadIdx.x;
  if (idx >= NTOK * D_INNER) return;
  const int d  = idx % D_INNER;
  const int bl = idx / D_INNER;
  const float z = xz[(size_t)bl * (2 * D_INNER) + D_INNER + d];
  const float y = ycore[idx] + Dp[d] * xc[idx];
  yact[idx] = y * (z / (1.f + __expf(-z)));
}

__global__ void resadd_kernel(float* __restrict__ x,
                              const float* __restrict__ p, int n) {
  const int idx = blockIdx.x * blockDim.x + threadIdx.x;
  if (idx < n) x[idx] += p[idx];
}

// ---------------------------------------------------------------------------
extern "C" void kernel_launch(void* const* d_in, const int* in_sizes, int n_in,
                              void* d_out, int out_size, void* d_ws,
                              size_t ws_size, hipStream_t stream) {
  const float* x_in      = (const float*)d_in[0];
  const float* in_proj_w = (const float*)d_in[1];
  const float* conv_w    = (const float*)d_in[2];
  const float* conv_b    = (const float*)d_in[3];
  const float* x_proj_w  = (const float*)d_in[4];
  const float* dt_proj_w = (const float*)d_in[5];
  const float* dt_proj_b = (const float*)d_in[6];
  const float* A_log     = (const float*)d_in[7];
  const float* D_vec     = (const float*)d_in[8];
  const float* out_proj_w= (const float*)d_in[9];
  const float* norm_w    = (const float*)d_in[10];
  const float* final_w   = (const float*)d_in[11];

  float* ws = (float*)d_ws;
  float* xcur  = ws;                       // 1024*768
  float* xn    = xcur  + NTOK * D_MODEL;   // 1024*768
  float* xz    = xn    + NTOK * D_MODEL;   // 1024*3072
  float* xcv   = xz    + NTOK * 2 * D_INNER;  // 1024*1536
  float* xdbl  = xcv   + NTOK * D_INNER;   // 1024*80
  float* dtb   = xdbl  + NTOK * XDBL_C;    // 1024*1536
  float* negA  = dtb   + NTOK * D_INNER;   // 1536*16
  float* ycore = negA  + D_INNER * D_STATE;// 1024*1536
  float* yact  = ycore + NTOK * D_INNER;   // 1024*1536
  float* proj  = yact  + NTOK * D_INNER;   // 1024*768

  hipMemcpyAsync(xcur, x_in, sizeof(float) * NTOK * D_MODEL,
                 hipMemcpyDeviceToDevice, stream);

  const int EW  = (NTOK * D_INNER + 255) / 256;  // 6144 blocks
  const int EWM = (NTOK * D_MODEL + 255) / 256;  // 3072 blocks
  const size_t SCAN_SMEM =
      (size_t)(2 * D_INNER * D_STATE + 16 * 256 + 256 + 256 + 32 + 8) *
      sizeof(float);

  for (int i = 0; i < N_LAYERS; ++i) {
    const float* inw  = in_proj_w  + (size_t)i * 2 * D_INNER * D_MODEL;
    const float* cw   = conv_w     + (size_t)i * D_INNER * D_CONV;
    const float* cb   = conv_b     + (size_t)i * D_INNER;
    const float* xpw  = x_proj_w   + (size_t)i * XDBL_C * D_INNER;
    const float* dtpw = dt_proj_w  + (size_t)i * D_INNER * DT_RANK;
    const float* dtpb = dt_proj_b  + (size_t)i * D_INNER;
    const float* alog = A_log      + (size_t)i * D_INNER * D_STATE;
    const float* dvec = D_vec      + (size_t)i * D_INNER;
    const float* outw = out_proj_w + (size_t)i * D_MODEL * D_INNER;
    const float* nw   = norm_w     + (size_t)i * D_MODEL;

    rmsnorm_kernel<<<NTOK, 256, 0, stream>>>(xcur, nw, xn, D_MODEL);

    // xz = xn @ in_w^T   (1024 x 3072, K=768)
    gemm_wmma_f32<<<dim3((2 * D_INNER) / 64, NTOK / 16), 128, 0, stream>>>(
        xn, D_MODEL, inw, D_MODEL, xz, 2 * D_INNER, 2 * D_INNER, D_MODEL);

    conv_silu_kernel<<<EW, 256, 0, stream>>>(xz, cw, cb, xcv);

    // x_dbl = xc @ xp_w^T   (1024 x 80, K=1536)
    gemm_wmma_f32<<<dim3((XDBL_C + 63) / 64, NTOK / 16), 128, 0, stream>>>(
        xcv, D_INNER, xpw, D_INNER, xdbl, XDBL_C, XDBL_C, D_INNER);

    // dt_raw = dt_lr @ dtp_w^T   (1024 x 1536, K=48, lda=80)
    gemm_wmma_f32<<<dim3(D_INNER / 64, NTOK / 16), 128, 0, stream>>>(
        xdbl, XDBL_C, dtpw, DT_RANK, dtb, D_INNER, D_INNER, DT_RANK);

    softplus_kernel<<<EW, 256, 0, stream>>>(dtb, dtpb);
    nega_kernel<<<(D_INNER * D_STATE + 255) / 256, 256, 0, stream>>>(alog, negA);

    scan_kernel<<<BATCH_N, SCAN_THREADS, SCAN_SMEM, stream>>>(dtb, xdbl, xcv,
                                                              negA, ycore);

    combine_kernel<<<EW, 256, 0, stream>>>(ycore, xcv, xz, dvec, yact);

    // proj = yact @ out_w^T   (1024 x 768, K=1536)
    gemm_wmma_f32<<<dim3(D_MODEL / 64, NTOK / 16), 128, 0, stream>>>(
        yact, D_INNER, outw, D_INNER, proj, D_MODEL, D_MODEL, D_INNER);

    resadd_kernel<<<EWM, 256, 0, stream>>>(xcur, proj, NTOK * D_MODEL);
  }

  rmsnorm_kernel<<<NTOK, 256, 0, stream>>>(xcur, final_w, (float*)d_out,
                                           D_MODEL);
}